// GraphTransformer_833223655848
// MI455X (gfx1250) — compile-verified
//
#include <hip/hip_runtime.h>
#include <math.h>

#define N_NODES 50000
#define N_EDGES 800000
#define CH 128   // all layer widths are 128

typedef __attribute__((ext_vector_type(2))) float v2f;
typedef __attribute__((ext_vector_type(8))) float v8f;

// ---------------- monotone float <-> uint key (for atomicMax segment max) ----
__device__ __forceinline__ unsigned f2key(float f) {
    unsigned u = __float_as_uint(f);
    return (u & 0x80000000u) ? ~u : (u | 0x80000000u);
}
__device__ __forceinline__ float key2f(unsigned k) {
    unsigned u = (k & 0x80000000u) ? (k & 0x7FFFFFFFu) : ~k;
    return __uint_as_float(u);
}

// ---------------- WMMA fp32 GEMM: Y[nrows,128] = X[nrows,128] @ W[128,128] + b
// block = 256 threads = 8 waves; block covers 16 rows; wave w covers cols [16w,16w+16)
__global__ __launch_bounds__(256) void gemm128_wmma(
    const float* __restrict__ X, const float* __restrict__ W,
    const float* __restrict__ bias, float* __restrict__ Y, int nrows)
{
    const int wave = threadIdx.x >> 5;          // 0..7 -> N tile
    const int lane = threadIdx.x & 31;
    const int m0   = blockIdx.x * 16;
    if (m0 >= nrows) return;
    const int n0   = wave * 16;
    const int r    = lane & 15;                 // A: row within tile / B,C: col
    const int half = lane >> 4;                 // 0 or 1
    const int kh   = half * 2;                  // K sub-offset held by this half-wave

    v8f acc = {};
    const float* Arow = X + (size_t)(m0 + r) * CH;   // this lane's A row
    const float* Bcol = W + n0 + r;                  // this lane's B column

    for (int kk = 0; kk < CH; kk += 4) {
        v2f a, b;
        a.x = Arow[kk + kh];
        a.y = Arow[kk + kh + 1];
        b.x = Bcol[(size_t)(kk + kh) * CH];
        b.y = Bcol[(size_t)(kk + kh + 1) * CH];
        acc = __builtin_amdgcn_wmma_f32_16x16x4_f32(
                  false, a, false, b, (short)0, acc, false, false);
    }

    const float bn = bias[n0 + r];
    const int mb = half * 8;
#pragma unroll
    for (int i = 0; i < 8; ++i) {
        Y[(size_t)(m0 + mb + i) * CH + n0 + r] = acc[i] + bn;
    }
}

// ---------------- softmax state init -----------------------------------------
__global__ void init_softmax(unsigned* __restrict__ mkey, float* __restrict__ denom, int n)
{
    int i = blockIdx.x * blockDim.x + threadIdx.x;
    if (i < n) { mkey[i] = 0u; denom[i] = 0.0f; }  // key 0 < key(-inf) < any finite
}

// ---------------- per-(edge,head): alpha = q[dst]·k[src]/sqrt(d), seg-max ----
__global__ void edge_alpha_kernel(
    const float* __restrict__ q, const float* __restrict__ k,
    const int* __restrict__ src, const int* __restrict__ dst,
    float* __restrict__ alpha, unsigned* __restrict__ mkey,
    int E, int H, int d, float inv_sqrt_d)
{
    int t = blockIdx.x * blockDim.x + threadIdx.x;
    if (t >= E * H) return;
    int e = t / H, h = t - e * H;
    int s = src[e], dd = dst[e];
    const float4* qp = (const float4*)(q + (size_t)dd * CH + h * d);
    const float4* kp = (const float4*)(k + (size_t)s  * CH + h * d);
    float acc = 0.0f;
    for (int j = 0; j < d / 4; ++j) {
        float4 a = qp[j], b = kp[j];
        acc += a.x * b.x + a.y * b.y + a.z * b.z + a.w * b.w;
    }
    float al = acc * inv_sqrt_d;
    alpha[t] = al;
    atomicMax(mkey + (size_t)dd * H + h, f2key(al));
}

// ---------------- e = exp(alpha - m[dst]);  denom[dst] += e ------------------
__global__ void edge_exp_kernel(
    float* __restrict__ alpha, const unsigned* __restrict__ mkey,
    float* __restrict__ denom, const int* __restrict__ dst, int E, int H)
{
    int t = blockIdx.x * blockDim.x + threadIdx.x;
    if (t >= E * H) return;
    int e = t / H, h = t - e * H;
    int dd = dst[e];
    float mm = key2f(mkey[(size_t)dd * H + h]);
    float ex = expf(alpha[t] - mm);
    alpha[t] = ex;
    atomicAdd(denom + (size_t)dd * H + h, ex);
}

// ---------------- out[dst,ch] += v[src,ch] * e/(denom[dst]+eps) --------------
__global__ void edge_agg_kernel(
    const float* __restrict__ v, const float* __restrict__ ealpha,
    const float* __restrict__ denom, const int* __restrict__ src,
    const int* __restrict__ dst, float* __restrict__ out, int E, int H, int d)
{
    int t = blockIdx.x * blockDim.x + threadIdx.x;   // E*128 <= 102.4M fits int
    if (t >= E * CH) return;
    int e = t >> 7, ch = t & (CH - 1);
    int h = ch / d;
    int s = src[e], dd = dst[e];
    float att = ealpha[(size_t)e * H + h] / (denom[(size_t)dd * H + h] + 1e-16f);
    atomicAdd(out + (size_t)dd * CH + ch, v[(size_t)s * CH + ch] * att);
}

// ---------------- elementwise ReLU -------------------------------------------
__global__ void relu_kernel(float* __restrict__ x, int n)
{
    int i = blockIdx.x * blockDim.x + threadIdx.x;
    if (i < n) x[i] = fmaxf(x[i], 0.0f);
}

// =============================================================================
extern "C" void kernel_launch(void* const* d_in, const int* in_sizes, int n_in,
                              void* d_out, int out_size, void* d_ws, size_t ws_size,
                              hipStream_t stream)
{
    const float* x    = (const float*)d_in[0];
    const int*   eidx = (const int*)d_in[1];
    const int*   src  = eidx;
    const int*   dst  = eidx + N_EDGES;
    const float* Wq1 = (const float*)d_in[2],  *bq1 = (const float*)d_in[3];
    const float* Wk1 = (const float*)d_in[4],  *bk1 = (const float*)d_in[5];
    const float* Wv1 = (const float*)d_in[6],  *bv1 = (const float*)d_in[7];
    const float* Ws1 = (const float*)d_in[8],  *bs1 = (const float*)d_in[9];
    const float* Wq2 = (const float*)d_in[10], *bq2 = (const float*)d_in[11];
    const float* Wk2 = (const float*)d_in[12], *bk2 = (const float*)d_in[13];
    const float* Wv2 = (const float*)d_in[14], *bv2 = (const float*)d_in[15];
    const float* Ws2 = (const float*)d_in[16], *bs2 = (const float*)d_in[17];
    float* out = (float*)d_out;

    // ---- workspace carve-up (floats) ----
    float* ws = (float*)d_ws;
    const size_t NC = (size_t)N_NODES * CH;
    float*    q     = ws;                      // N*128
    float*    k     = q + NC;                  // N*128
    float*    v     = k + NC;                  // N*128
    float*    h     = v + NC;                  // N*128  (layer-1 output)
    float*    alpha = h + NC;                  // E*4 (layer1), E (layer2)
    unsigned* mkey  = (unsigned*)(alpha + (size_t)N_EDGES * 4);  // N*4
    float*    denom = (float*)(mkey + (size_t)N_NODES * 4);      // N*4

    const dim3 B256(256);
    const int  gemm_grid = (N_NODES + 15) / 16;          // 3125
    const int  node_grid4 = (N_NODES * 4 + 255) / 256;
    const int  node_grid1 = (N_NODES + 255) / 256;
    const int  eh4_grid  = (N_EDGES * 4 + 255) / 256;
    const int  eh1_grid  = (N_EDGES + 255) / 256;
    const int  agg_grid  = (N_EDGES * CH + 255) / 256;   // 400000
    const int  relu_grid = (N_NODES * CH + 255) / 256;

    // ================= Layer 1 (H=4, d=32) =================
    gemm128_wmma<<<gemm_grid, B256, 0, stream>>>(x, Wq1, bq1, q, N_NODES);
    gemm128_wmma<<<gemm_grid, B256, 0, stream>>>(x, Wk1, bk1, k, N_NODES);
    gemm128_wmma<<<gemm_grid, B256, 0, stream>>>(x, Wv1, bv1, v, N_NODES);
    gemm128_wmma<<<gemm_grid, B256, 0, stream>>>(x, Ws1, bs1, h, N_NODES);  // skip term

    init_softmax<<<node_grid4, B256, 0, stream>>>(mkey, denom, N_NODES * 4);
    edge_alpha_kernel<<<eh4_grid, B256, 0, stream>>>(q, k, src, dst, alpha, mkey,
                                                     N_EDGES, 4, 32, 1.0f / sqrtf(32.0f));
    edge_exp_kernel<<<eh4_grid, B256, 0, stream>>>(alpha, mkey, denom, dst, N_EDGES, 4);
    edge_agg_kernel<<<agg_grid, B256, 0, stream>>>(v, alpha, denom, src, dst, h,
                                                   N_EDGES, 4, 32);
    relu_kernel<<<relu_grid, B256, 0, stream>>>(h, N_NODES * CH);

    // ================= Layer 2 (H=1, d=128) =================
    gemm128_wmma<<<gemm_grid, B256, 0, stream>>>(h, Wq2, bq2, q, N_NODES);
    gemm128_wmma<<<gemm_grid, B256, 0, stream>>>(h, Wk2, bk2, k, N_NODES);
    gemm128_wmma<<<gemm_grid, B256, 0, stream>>>(h, Wv2, bv2, v, N_NODES);
    gemm128_wmma<<<gemm_grid, B256, 0, stream>>>(h, Ws2, bs2, out, N_NODES); // skip term

    init_softmax<<<node_grid1, B256, 0, stream>>>(mkey, denom, N_NODES);
    edge_alpha_kernel<<<eh1_grid, B256, 0, stream>>>(q, k, src, dst, alpha, mkey,
                                                     N_EDGES, 1, 128, 1.0f / sqrtf(128.0f));
    edge_exp_kernel<<<eh1_grid, B256, 0, stream>>>(alpha, mkey, denom, dst, N_EDGES, 1);
    edge_agg_kernel<<<agg_grid, B256, 0, stream>>>(v, alpha, denom, src, dst, out,
                                                   N_EDGES, 1, 128);
}